// RNNLanguageModel_8658654068991
// MI455X (gfx1250) — compile-verified
//
#include <hip/hip_runtime.h>
#include <hip/hip_bf16.h>

typedef __attribute__((ext_vector_type(2))) float v2f;
typedef __attribute__((ext_vector_type(4))) float v4f;
typedef __attribute__((ext_vector_type(8))) float v8f;

#define VOCAB 96
#define EMBED 32
#define HIDDEN 64
#define G4 256          // 4*HIDDEN
#define NB 256          // batch
#define NT 1024         // time steps
#define NV 96           // vocab / logits cols

#define HS 68           // padded h row stride: 272B rows -> 16B aligned, banks (4r+c)&63 distinct
#define GS 264          // padded gates row stride: 1056B rows -> 16B aligned
#define LS 104          // padded logits row stride: 416B rows -> 16B aligned

static __device__ __forceinline__ v8f wmma_f32(v2f a, v2f b, v8f c) {
    // D = A(16x4 f32) * B(4x16 f32) + C(16x16 f32)
    return __builtin_amdgcn_wmma_f32_16x16x4_f32(
        /*neg_a=*/false, a, /*neg_b=*/false, b,
        /*c_mod=*/(short)0, c, /*reuse_a=*/false, /*reuse_b=*/false);
}

static __device__ __forceinline__ float sigmoidf_fast(float x) {
    return 1.0f / (1.0f + __expf(-x));
}

// -------- Kernel A: proj[v][g] = (b_ih+b_hh)[g] + sum_e emb[v][e]*W_ih[g][e];
//          also zero the loss accumulator slot. (96x256 table = 98KB, L2-resident)
__global__ void lstm_proj_init(const float* __restrict__ emb,
                               const float* __restrict__ W_ih,
                               const float* __restrict__ b_ih,
                               const float* __restrict__ b_hh,
                               float* __restrict__ proj,
                               float* __restrict__ loss_slot) {
    if (threadIdx.x == 0) *loss_slot = 0.0f;
    for (int e = threadIdx.x; e < VOCAB * G4; e += 256) {
        const int v = e / G4, g = e % G4;
        float acc = b_ih[g] + b_hh[g];
        const float* er = emb  + v * EMBED;
        const float* wr = W_ih + g * EMBED;
#pragma unroll
        for (int k = 0; k < EMBED; ++k) acc = fmaf(er[k], wr[k], acc);
        proj[e] = acc;
    }
}

// -------- Kernel B: fused LSTM scan + logits + log-softmax loss.
// grid = 16 blocks (16 batch rows each), 256 threads = 8 wave32s.
__global__ void __launch_bounds__(256, 1)
lstm_fused(const int*   __restrict__ idx,
           const int*   __restrict__ targets,
           const float* __restrict__ W_hh,    // [256][64] row-major
           const float* __restrict__ fc_w,    // [96][64] row-major
           const float* __restrict__ fc_b,    // [96]
           const float* __restrict__ proj,    // [96][256]
           float*       __restrict__ logits_out,  // [B*T*96]
           float*       __restrict__ loss_out) {
    __shared__ __align__(16) float h_lds[16 * HS];
    __shared__ __align__(16) float gates_lds[16 * GS];
    __shared__ __align__(16) float logits_lds[16 * LS];
    __shared__ float wsum[8];

    const int tid  = threadIdx.x;
    const int lane = tid & 31;
    const int wave = tid >> 5;
    const int half = lane >> 4;     // selects K pair / output M half
    const int l16  = lane & 15;
    const int b0   = blockIdx.x * 16;

    // zero initial hidden state
    for (int i = tid; i < 16 * HS; i += 256) h_lds[i] = 0.0f;

    // ---- preload W_hh^T B-fragments: this wave owns gate columns [wave*32, wave*32+32)
    v2f bfrag[2][16];
#pragma unroll
    for (int tt = 0; tt < 2; ++tt) {
        const int n = wave * 32 + tt * 16 + l16;           // 0..255
        const float* wrow = W_hh + n * HIDDEN;
#pragma unroll
        for (int kk = 0; kk < 16; ++kk) {
            const int k = kk * 4 + half * 2;
            v2f b; b.x = wrow[k]; b.y = wrow[k + 1];
            bfrag[tt][kk] = b;
        }
    }
    // ---- preload fc_w^T fragments: waves 0..5 own logit columns [wave*16, wave*16+16)
    v2f fcfrag[16];
    float fcb = 0.0f;
    if (wave < 6) {
        const int n = wave * 16 + l16;                     // 0..95
        const float* frow = fc_w + n * HIDDEN;
#pragma unroll
        for (int kk = 0; kk < 16; ++kk) {
            const int k = kk * 4 + half * 2;
            v2f b; b.x = frow[k]; b.y = frow[k + 1];
            fcfrag[kk] = b;
        }
        fcb = fc_b[n];
    }

    // elementwise-cell ownership: row em (0..15), hidden cols ej..ej+3
    const int em = tid >> 4;
    const int ej = (tid & 15) * 4;
    const int* idx_row = idx + (size_t)(b0 + em) * NT;
    float c_reg[4] = {0.f, 0.f, 0.f, 0.f};
    float loss_acc = 0.0f;

    // softmax row pointers for this wave (rows 2w, 2w+1)
    const int* tgt_row0 = targets + (size_t)(b0 + wave * 2 + 0) * NT;
    const int* tgt_row1 = targets + (size_t)(b0 + wave * 2 + 1) * NT;

    // logits store: 16x96 tile = 384 float4 chunks; thread does chunk tid and
    // (tid<128) chunk tid+256.  Base pointers hoisted; only +t*NV varies.
    const int cid0 = tid;
    const int m0   = cid0 / 24;
    const int c0   = (cid0 - m0 * 24) * 4;
    const float* lsrc0 = logits_lds + m0 * LS + c0;
    float*       gdst0 = logits_out + (size_t)(b0 + m0) * NT * NV + c0;
    const int cid1 = tid + 256;
    const int m1   = cid1 / 24;
    const int c1   = (cid1 - m1 * 24) * 4;
    const float* lsrc1 = logits_lds + m1 * LS + c1;
    float*       gdst1 = (tid < 128)
        ? logits_out + (size_t)(b0 + m1) * NT * NV + c1 : nullptr;

    __syncthreads();

    for (int t = 0; t < NT; ++t) {
        // ================= gates_raw = h @ W_hh^T  (16x64 x 64x256) =================
        v8f accA = {};
        v8f accB = {};
#pragma unroll
        for (int kk = 0; kk < 16; ++kk) {
            const int k = kk * 4 + half * 2;
            v2f a;
            a.x = h_lds[l16 * HS + k];
            a.y = h_lds[l16 * HS + k + 1];
            accA = wmma_f32(a, bfrag[0][kk], accA);
            accB = wmma_f32(a, bfrag[1][kk], accB);
        }
        {
            const int n0 = wave * 32 + l16;
            const int mr = half * 8;
#pragma unroll
            for (int j = 0; j < 8; ++j) {
                gates_lds[(mr + j) * GS + n0]      = accA[j];
                gates_lds[(mr + j) * GS + n0 + 16] = accB[j];
            }
        }
        __syncthreads();   // barrier 1: gates visible; h_lds reads of iter t done

        // ================= LSTM cell (elementwise, fully 128-bit) =================
        {
            const int token = idx_row[t];
            const float* prow = proj + token * G4;
            if (t + 1 < NT) {
                const int ntok = idx_row[t + 1];
                __builtin_prefetch(proj + ntok * G4 + ej, 0, 3);  // global_prefetch_b8
            }
            const float* grow = gates_lds + em * GS;
            const v4f gi4 = *(const v4f*)(grow + ej)        + *(const v4f*)(prow + ej);
            const v4f gf4 = *(const v4f*)(grow + 64 + ej)   + *(const v4f*)(prow + 64 + ej);
            const v4f gg4 = *(const v4f*)(grow + 128 + ej)  + *(const v4f*)(prow + 128 + ej);
            const v4f go4 = *(const v4f*)(grow + 192 + ej)  + *(const v4f*)(prow + 192 + ej);
            v4f hv;
#pragma unroll
            for (int jj = 0; jj < 4; ++jj) {
                const float gi = sigmoidf_fast(gi4[jj]);
                const float gf = sigmoidf_fast(gf4[jj]);
                const float gg = tanhf(gg4[jj]);
                const float go = sigmoidf_fast(go4[jj]);
                const float c  = fmaf(gf, c_reg[jj], gi * gg);
                c_reg[jj] = c;
                hv[jj] = go * tanhf(c);
            }
            *(v4f*)(h_lds + em * HS + ej) = hv;
        }
        __syncthreads();   // barrier 2: new h visible

        // ================= logits = h @ fc_w^T + fc_b  (16x64 x 64x96) =================
        if (wave < 6) {
            v8f lacc = {};
#pragma unroll
            for (int kk = 0; kk < 16; ++kk) {
                const int k = kk * 4 + half * 2;
                v2f a;
                a.x = h_lds[l16 * HS + k];
                a.y = h_lds[l16 * HS + k + 1];
                lacc = wmma_f32(a, fcfrag[kk], lacc);
            }
            const int n  = wave * 16 + l16;
            const int mr = half * 8;
#pragma unroll
            for (int j = 0; j < 8; ++j)
                logits_lds[(mr + j) * LS + n] = lacc[j] + fcb;
        }
        __syncthreads();   // barrier 3: logits tile visible

        // ================= log-softmax NLL (wave handles rows 2w, 2w+1) =================
#pragma unroll
        for (int rr = 0; rr < 2; ++rr) {
            const int m = wave * 2 + rr;
            const float v0 = logits_lds[m * LS + lane];
            const float v1 = logits_lds[m * LS + lane + 32];
            const float v2 = logits_lds[m * LS + lane + 64];
            float mx = fmaxf(v0, fmaxf(v1, v2));
#pragma unroll
            for (int off = 16; off > 0; off >>= 1)
                mx = fmaxf(mx, __shfl_xor(mx, off, 32));
            float se = __expf(v0 - mx) + __expf(v1 - mx) + __expf(v2 - mx);
#pragma unroll
            for (int off = 16; off > 0; off >>= 1)
                se += __shfl_xor(se, off, 32);
            if (lane == 0) {
                const int tg = (rr == 0) ? tgt_row0[t] : tgt_row1[t];
                const float lt = logits_lds[m * LS + tg];
                loss_acc += (mx + __logf(se)) - lt;   // -logp[target]
            }
        }

        // ================= stream logits tile to HBM as b128 =================
        *(v4f*)(gdst0 + (size_t)t * NV) = *(const v4f*)lsrc0;
        if (gdst1) *(v4f*)(gdst1 + (size_t)t * NV) = *(const v4f*)lsrc1;

        // No trailing barrier needed: logits_lds/h_lds reads above complete
        // before this wave reaches barrier 1 of iteration t+1, and their next
        // writers only run after barriers 2/1 of t+1 respectively.
    }

    // ---- loss reduction: only lane0 threads carry loss
    if (lane == 0) wsum[wave] = loss_acc;
    __syncthreads();
    if (tid == 0) {
        float s = 0.0f;
#pragma unroll
        for (int w = 0; w < 8; ++w) s += wsum[w];
        atomicAdd(loss_out, s * (1.0f / ((float)NB * (float)NT)));
    }
}

extern "C" void kernel_launch(void* const* d_in, const int* in_sizes, int n_in,
                              void* d_out, int out_size, void* d_ws, size_t ws_size,
                              hipStream_t stream) {
    // setup_inputs order:
    // 0 idx[i32 B*T], 1 targets[i32 B*T], 2 embedding[f32 96*32],
    // 3 W_ih[f32 256*32], 4 W_hh[f32 256*64], 5 b_ih[f32 256],
    // 6 b_hh[f32 256], 7 fc_w[f32 96*64], 8 fc_b[f32 96]
    const int*   idx     = (const int*)  d_in[0];
    const int*   targets = (const int*)  d_in[1];
    const float* emb     = (const float*)d_in[2];
    const float* W_ih    = (const float*)d_in[3];
    const float* W_hh    = (const float*)d_in[4];
    const float* b_ih    = (const float*)d_in[5];
    const float* b_hh    = (const float*)d_in[6];
    const float* fc_w    = (const float*)d_in[7];
    const float* fc_b    = (const float*)d_in[8];

    float* out    = (float*)d_out;
    float* logits = out;                                   // [B*T*96]
    float* loss   = out + (size_t)NB * NT * NV;            // scalar slot
    float* proj   = (float*)d_ws;                          // [96][256] = 98304 B

    (void)in_sizes; (void)n_in; (void)out_size; (void)ws_size;

    lstm_proj_init<<<1, 256, 0, stream>>>(emb, W_ih, b_ih, b_hh, proj, loss);
    lstm_fused<<<NB / 16, 256, 0, stream>>>(idx, targets, W_hh, fc_w, fc_b,
                                            proj, logits, loss);
}